// rate_RNN_mante_26645977104955
// MI455X (gfx1250) — compile-verified
//
#include <hip/hip_runtime.h>
#include <cmath>

typedef __attribute__((ext_vector_type(2))) float v2f;
typedef __attribute__((ext_vector_type(8))) float v8f;

// exp(-DT/TAUM) = exp(-0.05)
#define LM   0.95122942450071403f
#define OMLM 0.04877057549928597f
#define TWO_LOG2E 2.8853900817779268f   // 2*log2(e)

constexpr int T_ = 1000, B_ = 64, I_ = 4, H_ = 1024, O_ = 2, P_ = 16;
constexpr int MEM_STRIDE = H_ + 4;   // 1028: row stride = 4 banks mod 64 -> conflict-free A-frags
constexpr int S_STRIDE   = 20;       // 20: conflict-free s-fragment loads

__device__ __forceinline__ float fast_exp2(float x) {
#if __has_builtin(__builtin_amdgcn_exp2f)
    return __builtin_amdgcn_exp2f(x);      // v_exp_f32
#else
    return exp2f(x);
#endif
}

// branch-free tanh: 1 - 2/(exp(2x)+1); clamp keeps exp finite (tanh saturates ~|x|>9)
__device__ __forceinline__ float fast_tanh(float v) {
    v = __builtin_fminf(__builtin_fmaxf(v, -10.0f), 10.0f);
    float e = fast_exp2(v * TWO_LOG2E);
    float r = __builtin_amdgcn_rcpf(e + 1.0f);   // v_rcp_f32
    return 1.0f - 2.0f * r;
}

__global__ __launch_bounds__(1024)
void rate_rnn_lowrank_kernel(const float* __restrict__ x,     // (T,B,I)
                             const float* __restrict__ Win,   // (H,I)
                             const float* __restrict__ Wout,  // (O,H)
                             const float* __restrict__ pin,   // (H,P)
                             const float* __restrict__ pout,  // (H,P)
                             const float* __restrict__ lvec,  // (P)
                             float* __restrict__ y)           // (T,B,O)
{
    __shared__ float mem_s[B_ * MEM_STRIDE];  // 263168 B : state in WMMA-friendly padded layout
    __shared__ float s_s[B_ * S_STRIDE];      //   5120 B : s = tanh(mem) @ pout
    __shared__ float y_s[B_ * 16];            //   4096 B : y partials (O padded to 16)
    __shared__ float x_s[B_ * I_];            //   1024 B : x_t staging

    const int tid  = threadIdx.x;
    const int lane = tid & 31;
    const int wv   = tid >> 5;      // 0..31 waves
    const int half = lane >> 4;     // 0/1 (K split across half-waves)
    const int l16  = lane & 15;

    // phase A: wave -> (M tile, K chunk of 128) ; phase B: wave -> (M tile, 8 strided N tiles)
    const int aM = wv >> 3;
    const int aK = wv & 7;
    const int bM = wv >> 3;
    const int bN = wv & 7;

    // Wout fragment: clamp row so the load is always in-bounds (no EXEC-guarded loads),
    // zero out lanes beyond O with a multiplicative mask.
    const int   wrow  = (l16 < O_) ? l16 : 0;
    const float wmask = (l16 < O_) ? 1.0f : 0.0f;
    const float* wout_row = &Wout[wrow * H_];

    // hoist lvec B-fragment scales (time-invariant): lv[kk] = (l[4kk+2*half], l[4kk+2*half+1])
    v2f lv[4];
#pragma unroll
    for (int kk = 0; kk < 4; ++kk) {
        const int p = kk * 4 + 2 * half;
        lv[kk].x = lvec[p];
        lv[kk].y = lvec[p + 1];
    }

    // mem_{-1} = 0
    for (int i = tid; i < B_ * MEM_STRIDE; i += 1024) mem_s[i] = 0.0f;
    __syncthreads();

#pragma unroll 1
    for (int t = 0; t <= T_; ++t) {
        // ---- zero reduction buffers, stage x_t, prefetch x_{t+1} ----
        for (int i = tid; i < B_ * S_STRIDE; i += 1024) s_s[i] = 0.0f;
        for (int i = tid; i < B_ * 16; i += 1024)       y_s[i] = 0.0f;
        if (t < T_ && tid < B_ * I_) x_s[tid] = x[t * B_ * I_ + tid];
        if (t + 1 < T_ && tid == 0)  __builtin_prefetch(&x[(t + 1) * B_ * I_], 0, 1);
        __syncthreads();

        // ---- Phase A: accS = tanh(mem) @ pout (K-split), accY = tanh(mem) @ Wout.T ----
        {
            v8f accS = {};
            v8f accY = {};
            const int row = aM * 16 + l16;        // A-frag: lane row = M
            const int k0  = aK * 128;
            const float* lds_row = &mem_s[row * MEM_STRIDE];
#pragma unroll 8
            for (int kk = 0; kk < 32; ++kk) {
                const int k = k0 + kk * 4 + 2 * half;        // this lane's K pair
                const v2f m2 = *(const v2f*)&lds_row[k];     // one ds_load_b64
                v2f a;
                a.x = fast_tanh(m2.x);
                a.y = fast_tanh(m2.y);
                v2f bp, bw;
                bp.x = pout[(k + 0) * P_ + l16];             // B-frag: lane col = N
                bp.y = pout[(k + 1) * P_ + l16];
                bw.x = wmask * wout_row[k + 0];              // unconditional load + mask
                bw.y = wmask * wout_row[k + 1];
                accS = __builtin_amdgcn_wmma_f32_16x16x4_f32(
                           false, a, false, bp, (short)0, accS, false, false);
                accY = __builtin_amdgcn_wmma_f32_16x16x4_f32(
                           false, a, false, bw, (short)0, accY, false, false);
            }
            // reduce K-chunk partials across the 8 waves sharing this M tile
#pragma unroll
            for (int j = 0; j < 8; ++j) {
                const int b = aM * 16 + j + 8 * half;   // C-frag row
                atomicAdd(&s_s[b * S_STRIDE + l16], accS[j]);
                if (l16 < O_) atomicAdd(&y_s[b * 16 + l16], accY[j]);
            }
        }
        __syncthreads();

        // ---- write y_{t-1} = Wout @ tanh(mem_{t-1}) ----
        if (t > 0 && tid < B_ * O_) {
            y[(t - 1) * B_ * O_ + tid] = y_s[(tid >> 1) * 16 + (tid & 1)];
        }

        // ---- Phase B: rec = s @ (l*pin).T ; leaky update of mem ----
        if (t < T_) {
            const int srow = bM * 16 + l16;
            const float* s_row = &s_s[srow * S_STRIDE];
#pragma unroll 1
            for (int i = 0; i < 8; ++i) {
                const int nT   = bN + 8 * i;            // 0..63
                const int hcol = nT * 16 + l16;         // this lane's H column
                v8f c = {};
#pragma unroll
                for (int kk = 0; kk < 4; ++kk) {
                    const int p = kk * 4 + 2 * half;
                    const v2f a = *(const v2f*)&s_row[p];    // one ds_load_b64
                    v2f b;
                    b.x = pin[hcol * P_ + p]     * lv[kk].x;
                    b.y = pin[hcol * P_ + p + 1] * lv[kk].y;
                    c = __builtin_amdgcn_wmma_f32_16x16x4_f32(
                            false, a, false, b, (short)0, c, false, false);
                }
                const float4 w = *(const float4*)&Win[hcol * I_];
#pragma unroll
                for (int j = 0; j < 8; ++j) {
                    const int b_ = bM * 16 + j + 8 * half;   // C-frag row = batch
                    const float4 xv = *(const float4*)&x_s[b_ * I_];
                    const float iin = w.x * xv.x + w.y * xv.y + w.z * xv.z + w.w * xv.w;
                    const float m_old = mem_s[b_ * MEM_STRIDE + hcol];
                    mem_s[b_ * MEM_STRIDE + hcol] = LM * m_old + OMLM * (iin + c[j]);
                }
            }
        }
        __syncthreads();
    }
}

extern "C" void kernel_launch(void* const* d_in, const int* in_sizes, int n_in,
                              void* d_out, int out_size, void* d_ws, size_t ws_size,
                              hipStream_t stream) {
    const float* x    = (const float*)d_in[0];
    const float* Win  = (const float*)d_in[1];
    const float* Wout = (const float*)d_in[2];
    const float* pin  = (const float*)d_in[3];
    const float* pout = (const float*)d_in[4];
    const float* lvec = (const float*)d_in[5];
    float* y = (float*)d_out;

    rate_rnn_lowrank_kernel<<<1, 1024, 0, stream>>>(x, Win, Wout, pin, pout, lvec, y);
}